// HGNN_944892805250
// MI455X (gfx1250) — compile-verified
//
#include <hip/hip_runtime.h>
#include <stdint.h>

// Problem sizes (match reference)
#define N_NODES 100000
#define N_EDGES 50000
#define M_INC   1600000
#define D_HID   64

#define TILE 16            // incidences per block (16 groups x 16 lanes = 256 thr)

#define GLOBAL_AS __attribute__((address_space(1)))
#define LDS_AS    __attribute__((address_space(3)))

#if defined(__has_builtin)
#if __has_builtin(__builtin_amdgcn_global_load_async_to_lds_b32)
#define HAVE_ASYNC_LDS 1
#endif
#endif

#if defined(__has_builtin)
#if __has_builtin(__builtin_amdgcn_s_wait_asynccnt)
#define WAIT_ASYNC0() __builtin_amdgcn_s_wait_asynccnt(0)
#endif
#endif
#ifndef WAIT_ASYNC0
#define WAIT_ASYNC0() asm volatile("s_wait_asynccnt 0" ::: "memory")
#endif

__device__ __forceinline__ void atomic_add_f32(float* p, float v) {
  // relaxed, agent scope -> no-return global_atomic_add_f32 on gfx1250
  (void)__hip_atomic_fetch_add(p, v, __ATOMIC_RELAXED, __HIP_MEMORY_SCOPE_AGENT);
}

// Stage one 32-bit word from global into LDS. On gfx1250 use the async
// global->LDS DMA path (ASYNCcnt-tracked); otherwise plain load + ds store.
__device__ __forceinline__ void stage_b32(const int* g, int* lds_generic) {
#if defined(HAVE_ASYNC_LDS)
  GLOBAL_AS int* gp = (GLOBAL_AS int*)(uintptr_t)g;
  // low 32 bits of the flat shared pointer are the LDS byte offset
  LDS_AS int* lp = (LDS_AS int*)(uint32_t)(uintptr_t)lds_generic;
  __builtin_amdgcn_global_load_async_to_lds_b32(gp, lp, 0, 0);
#else
  *lds_generic = *g;
#endif
}

__global__ void __launch_bounds__(256) zero_f32(float* __restrict__ p, int n4) {
  int i = blockIdx.x * blockDim.x + threadIdx.x;
  if (i < n4) {
    float4 z; z.x = 0.f; z.y = 0.f; z.z = 0.f; z.w = 0.f;
    reinterpret_cast<float4*>(p)[i] = z;
  }
}

// Dn[n] += w[e]; Be[e] += 1 for each incidence
__global__ void __launch_bounds__(256) degrees_kernel(
    const int* __restrict__ node_idx, const int* __restrict__ edge_idx,
    const float* __restrict__ ew, float* __restrict__ Dn, float* __restrict__ Be) {
  int m = blockIdx.x * blockDim.x + threadIdx.x;
  if (m >= M_INC) return;
  int n = node_idx[m];
  int e = edge_idx[m];
  atomic_add_f32(&Dn[n], ew[e]);
  atomic_add_f32(&Be[e], 1.0f);
}

// in-place safe reciprocal: 1/v with v<=0 -> 0
__global__ void __launch_bounds__(256) invert_kernel(float* __restrict__ Dn,
                                                     float* __restrict__ Be) {
  int i = blockIdx.x * blockDim.x + threadIdx.x;
  if (i < N_NODES) { float v = Dn[i]; Dn[i] = (v > 0.f) ? (1.0f / v) : 0.0f; }
  if (i < N_EDGES) { float v = Be[i]; Be[i] = (v > 0.f) ? (1.0f / v) : 0.0f; }
}

// Generic incidence scatter: dst[didx[m]][:] += src[sidx[m]][:]
// Block handles TILE incidences; 16 lanes per incidence, float4 each (64 f32/row).
__global__ void __launch_bounds__(256) scatter_kernel(
    const float* __restrict__ src, const int* __restrict__ sidx,
    const int* __restrict__ didx, float* __restrict__ dst) {
  __shared__ int sh_idx[2 * TILE];  // [0..15]=src rows, [16..31]=dst rows
  const int tid  = threadIdx.x;
  const int base = blockIdx.x * TILE;

  // wave 0 (lanes 0..31, full EXEC) DMAs the 32 indices for this tile into LDS
  if (tid < 2 * TILE) {
    const int* g = (tid < TILE) ? (sidx + base + tid)
                                : (didx + base + (tid - TILE));
    stage_b32(g, &sh_idx[tid]);
  }
#if defined(HAVE_ASYNC_LDS)
  WAIT_ASYNC0();
#endif
  __syncthreads();

  const int grp = tid >> 4;   // incidence within tile
  const int ln  = tid & 15;   // float4 slot within row
  const int srow = sh_idx[grp];
  const int drow = sh_idx[TILE + grp];

  const float* sp = src + srow * D_HID + ln * 4;
  float*       dp = dst + drow * D_HID + ln * 4;
  // Warm the two 128B destination lines of this group's RMW target in GL2:
  // one global_prefetch_b8 from lane 0 and lane 8 of each 16-lane group.
  if ((tid & 7) == 0) __builtin_prefetch(dp, 0, 0);

  const float4 v = *reinterpret_cast<const float4*>(sp);
  atomic_add_f32(dp + 0, v.x);
  atomic_add_f32(dp + 1, v.y);
  atomic_add_f32(dp + 2, v.z);
  atomic_add_f32(dp + 3, v.w);
}

// edge_feat[e][:] *= Be_inv[e]
__global__ void __launch_bounds__(256) scale_kernel(float* __restrict__ ef,
                                                    const float* __restrict__ be_inv) {
  int i = blockIdx.x * blockDim.x + threadIdx.x;  // over N_EDGES*16 float4
  if (i >= N_EDGES * (D_HID / 4)) return;
  float s = be_inv[i >> 4];
  float4 v = reinterpret_cast<float4*>(ef)[i];
  v.x *= s; v.y *= s; v.z *= s; v.w *= s;
  reinterpret_cast<float4*>(ef)[i] = v;
}

// out = 0.5*(x + Dn_inv * acc), acc currently in out
__global__ void __launch_bounds__(256) finalize_kernel(const float* __restrict__ x,
                                                       const float* __restrict__ dn_inv,
                                                       float* __restrict__ out) {
  int i = blockIdx.x * blockDim.x + threadIdx.x;  // over N_NODES*16 float4
  if (i >= N_NODES * (D_HID / 4)) return;
  float s = dn_inv[i >> 4] * 0.5f;
  float4 a = reinterpret_cast<const float4*>(x)[i];
  float4 b = reinterpret_cast<float4*>(out)[i];
  b.x = 0.5f * a.x + s * b.x;
  b.y = 0.5f * a.y + s * b.y;
  b.z = 0.5f * a.z + s * b.z;
  b.w = 0.5f * a.w + s * b.w;
  reinterpret_cast<float4*>(out)[i] = b;
}

extern "C" void kernel_launch(void* const* d_in, const int* in_sizes, int n_in,
                              void* d_out, int out_size, void* d_ws, size_t ws_size,
                              hipStream_t stream) {
  const float* x        = (const float*)d_in[0];
  const int*   node_idx = (const int*)d_in[1];
  const int*   edge_idx = (const int*)d_in[2];
  const float* ew       = (const float*)d_in[3];
  float*       out      = (float*)d_out;

  // workspace layout (all fit comfortably in L2: 13.4 MB total)
  float* Dn = (float*)d_ws;             // N_NODES  node-degree -> reciprocal
  float* Be = Dn + N_NODES;             // N_EDGES  edge-cardinality -> reciprocal
  float* EF = Be + N_EDGES;             // N_EDGES * D_HID edge features

  const int ws_f4  = (N_NODES + N_EDGES + N_EDGES * D_HID) / 4;  // 837500
  const int out_f4 = (N_NODES * D_HID) / 4;                      // 1600000

  hipLaunchKernelGGL(zero_f32, dim3((ws_f4 + 255) / 256), dim3(256), 0, stream,
                     Dn, ws_f4);
  hipLaunchKernelGGL(zero_f32, dim3((out_f4 + 255) / 256), dim3(256), 0, stream,
                     out, out_f4);

  hipLaunchKernelGGL(degrees_kernel, dim3((M_INC + 255) / 256), dim3(256), 0, stream,
                     node_idx, edge_idx, ew, Dn, Be);
  hipLaunchKernelGGL(invert_kernel, dim3((N_NODES + 255) / 256), dim3(256), 0, stream,
                     Dn, Be);

  // stage 1: nodes -> hyperedges (raw sums), then scale by Be^-1
  hipLaunchKernelGGL(scatter_kernel, dim3(M_INC / TILE), dim3(256), 0, stream,
                     x, node_idx, edge_idx, EF);
  hipLaunchKernelGGL(scale_kernel, dim3((N_EDGES * (D_HID / 4) + 255) / 256),
                     dim3(256), 0, stream, EF, Be);

  // stage 2: hyperedges -> nodes (accumulate into out)
  hipLaunchKernelGGL(scatter_kernel, dim3(M_INC / TILE), dim3(256), 0, stream,
                     EF, edge_idx, node_idx, out);

  // out = (x + Dn^-1 * acc) / 2
  hipLaunchKernelGGL(finalize_kernel, dim3((N_NODES * (D_HID / 4) + 255) / 256),
                     dim3(256), 0, stream, x, Dn, out);
}